// StreamingComplexFSMN_61357902790815
// MI455X (gfx1250) — compile-verified
//
#include <hip/hip_runtime.h>

typedef float v2f __attribute__((ext_vector_type(2)));
typedef float v8f __attribute__((ext_vector_type(8)));

#define BDIM   32
#define TDIM   2000
#define DDIM   768
#define MTAPS  10
#define TILE   16
#define NCHUNK 5
#define CHUNK_TILES 25                 // 125 tiles / 5 chunks (odd -> pair loop + tail)
#define RING   32
#define CGROUPS (DDIM / 16)
#define STRIPS  (BDIM * CGROUPS * NCHUNK)

// One wave per strip (batch b, 16-channel group, T-chunk). Per 16-step tile:
//   async-copy 16 rows global->LDS ring, then D = A_toeplitz x X_window with
//   seven chained V_WMMA_F32_16X16X4_F32 (identity folded in as the lag-0
//   diagonal = 1.0), then 16x16 fp32 tile stored non-temporally.
__global__ __launch_bounds__(32)
void fsmn_fir_wmma_kernel(const float* __restrict__ xr,
                          const float* __restrict__ xi,
                          const float* __restrict__ wr,
                          const float* __restrict__ wi,
                          float* __restrict__ outbase)
{
    __shared__ __align__(16) float smem[RING][16];

    const int lane   = threadIdx.x;                  // wave32 lane
    int id           = blockIdx.x;
    const int tensor = id / STRIPS;
    id              -= tensor * STRIPS;
    const int chunk  = id % NCHUNK;
    const int cg     = (id / NCHUNK) % CGROUPS;
    const int b      = id / (NCHUNK * CGROUPS);

    const float* x = tensor ? xi : xr;
    const float* w = tensor ? wi : wr;
    float* out = outbase + (size_t)tensor * BDIM * TDIM * DDIM;

    const int d0  = cg * 16;
    const int ct0 = chunk * (CHUNK_TILES * TILE);
    const int ph  = ct0 & (RING - 1);                // ring phase of chunk start

    const float* xb = x   + (size_t)b * TDIM * DDIM + d0;
    float*       ob = out + (size_t)b * TDIM * DDIM + d0;

    // ---- taps ----
    float wt[MTAPS];
    #pragma unroll
    for (int k = 0; k < MTAPS; ++k) wt[k] = w[k];

    // ---- loop-invariant A: banded Toeplitz, identity folded in at lag 0 ----
    // A 16x4 f32: lanes 0-15: M=lane, K={0,1}; lanes 16-31: M=lane-16, K={2,3}
    const int am    = lane & 15;
    const int kbase = (lane >> 4) << 1;              // 0 or 2
    v2f A[7];
    #pragma unroll
    for (int j = 0; j < 7; ++j) {
        #pragma unroll
        for (int v = 0; v < 2; ++v) {
            int off = 4 * j + kbase + v;             // window offset 0..27
            int lag = am + MTAPS - off;              // lag 0 => identity
            float a = 0.0f;
            if (lag == 0) a = 1.0f;
            else if (lag >= 1 && lag <= MTAPS) a = wt[lag - 1];
            A[j][v] = a;
        }
    }

    // C/D 16x16 f32: vgpr r -> M=r (lanes 0-15) / M=r+8 (lanes 16-31); N=lane&15
    const int cn     = lane & 15;
    const int cm_add = (lane >> 4) * 8;

    // ---- causal halo rows ct0-10..ct0-1 (zeros below t=0) ----
    // slot(t) = (t & 31); for t = ct0-10+ro this is (ph + 22 + ro) & 31
    #pragma unroll
    for (int i = 0; i < 5; ++i) {                    // 10 rows * 16 ch = 160
        int idx = i * 32 + lane;
        int ro  = idx >> 4;
        int ch  = idx & 15;
        int t   = ct0 - MTAPS + ro;
        float v = (t >= 0) ? xb[(size_t)t * DDIM + ch] : 0.0f;
        smem[(ph + 22 + ro) & (RING - 1)][ch] = v;
    }

    // per-lane streaming pointers
    const int lr  = lane >> 2;                       // loader row 0..7
    const int lc4 = (lane & 3) * 4;                  // loader chan base (float4)
    const float* ga = xb + ((size_t)ct0 + lr) * DDIM + lc4;       // async src
    float*       gy = ob + ((size_t)ct0 + cm_add) * DDIM + cn;    // store dst

    auto tile_body = [&](int phase) {
        // prior tile's LDS reads retired before its slots are recycled
        asm volatile("s_wait_dscnt 0" ::: "memory");

        // async copy rows t0..t0+15 (2 x 512B) global -> LDS ring, streamed NT
        unsigned lds0 = (unsigned)(size_t)&smem[(phase + lr) & (RING - 1)][lc4];
        unsigned lds1 = (unsigned)(size_t)&smem[(phase + 8 + lr) & (RING - 1)][lc4];
        unsigned long long ga64 = (unsigned long long)(size_t)ga;
        asm volatile("global_load_async_to_lds_b128 %0, %1, off th:TH_LOAD_NT"
                     :: "v"(lds0), "v"(ga64) : "memory");
        asm volatile("global_load_async_to_lds_b128 %0, %1, off offset:24576 th:TH_LOAD_NT"
                     :: "v"(lds1), "v"(ga64) : "memory");   // +8 rows (8*768*4 B)
        asm volatile("s_wait_asynccnt 0" ::: "memory");

        // seven chained fp32 WMMAs over window rows t0-10..t0+15
        v8f c = {};
        #pragma unroll
        for (int j = 0; j < 7; ++j) {
            // B 4x16 f32 (mirrors A): vgpr0 rows {0,2}, vgpr1 rows {1,3}
            v2f bmat;
            #pragma unroll
            for (int v = 0; v < 2; ++v) {
                int off = 4 * j + kbase + v;
                // slot of row (t0 - 10 + off) == (phase + 22 + off) & 31
                bmat[v] = smem[(phase + 22 + off) & (RING - 1)][cn];
            }
            c = __builtin_amdgcn_wmma_f32_16x16x4_f32(
                    false, A[j], false, bmat, (short)0, c, false, false);
        }

        // store 16x16 tile, non-temporal (output never re-read)
        #pragma unroll
        for (int r = 0; r < 8; ++r)
            __builtin_nontemporal_store(c[r], gy + (size_t)r * DDIM);

        ga += (size_t)TILE * DDIM;
        gy += (size_t)TILE * DDIM;
    };

    // pair loop: ring phase alternates ph / ph^16 and is loop-invariant per slot
    for (int tp = 0; tp < CHUNK_TILES - 1; tp += 2) {
        tile_body(ph);
        tile_body(ph ^ 16);
    }
    tile_body(ph);    // tile 24: even parity -> phase ph
}

extern "C" void kernel_launch(void* const* d_in, const int* in_sizes, int n_in,
                              void* d_out, int out_size, void* d_ws, size_t ws_size,
                              hipStream_t stream) {
    const float* real = (const float*)d_in[0];
    const float* imag = (const float*)d_in[1];
    const float* wr   = (const float*)d_in[2];
    const float* wi   = (const float*)d_in[3];

    dim3 grid(2 * STRIPS);     // 2 * 32 * 48 * 5 = 15360 single-wave workgroups
    dim3 block(32);
    fsmn_fir_wmma_kernel<<<grid, block, 0, stream>>>(real, imag, wr, wi, (float*)d_out);
}